// TokenEmbedding_11484742550035
// MI455X (gfx1250) — compile-verified
//
#include <hip/hip_runtime.h>

// Reference shapes / constants (compile-time)
#define BATCH 32
#define SEQ   8192
#define CIN   7
#define KS    8       // KSIZE = M+1
#define TAU   3
#define PADL  21      // M * TAU
#define WOUT  2736    // (SEQ + PADL - KS)/TAU + 1
#define NKER  74
#define OUTCH 512     // (NKER-1)*CIN + 1

#define NTILES 5          // ceil(80/16) covering n rows 0..79 (74 valid)
#define WTILES 171        // 2736 / 16
#define TILES_PER_WAVE 9  // 171 = 9 * 19
#define WGROUPS 19
#define WAVES_TOTAL (BATCH * CIN * NTILES * WGROUPS)  // 21280
#define WAVES_PER_BLOCK 8

typedef float v2f __attribute__((ext_vector_type(2)));
typedef float v8f __attribute__((ext_vector_type(8)));

// Clamped, branch-free load of x[b, s, c] (xb pre-offset to x + b*SEQ*CIN + c)
__device__ __forceinline__ float ldx(const float* __restrict__ xb, int s) {
    const bool ok = (unsigned)s < (unsigned)SEQ;
    const float v = xb[(ok ? s : 0) * CIN];
    return ok ? v : 0.0f;
}

__global__ __launch_bounds__(256)
void rocket_dilconv_wmma(const float* __restrict__ x,
                         const float* __restrict__ kern,
                         float* __restrict__ out) {
    const int wave = blockIdx.x * WAVES_PER_BLOCK + (threadIdx.x >> 5);
    const int lane = threadIdx.x & 31;
    const int half = lane >> 4;   // 0: lanes 0-15, 1: lanes 16-31
    const int l15  = lane & 15;

    int t = wave;
    const int wg = t % WGROUPS; t /= WGROUPS;
    const int nt = t % NTILES;  t /= NTILES;
    const int c  = t % CIN;     t /= CIN;
    const int b  = t;           // < 32

    const int n0  = nt * 16;
    const int m   = n0 + l15;       // A-matrix row = kernel index
    const int kb0 = half * 2;       // K pair for WMMA #1 (K = 0..3)
    const int kb1 = 4 + half * 2;   // K pair for WMMA #2 (K = 4..7)

    // A tiles (16x4 f32): lane l15 = M, half selects which K pair this lane holds
    v2f a0 = {0.0f, 0.0f}, a1 = {0.0f, 0.0f};
    if (m < NKER) {
        const float* kr = kern + m * KS;
        a0.x = kr[kb0]; a0.y = kr[kb0 + 1];
        a1.x = kr[kb1]; a1.y = kr[kb1 + 1];
    }

    const float* __restrict__ xb = x + (size_t)b * SEQ * CIN + c;
    float* __restrict__ ob = out + (size_t)b * OUTCH * WOUT;

    for (int i = 0; i < TILES_PER_WAVE; ++i) {
        const int w0 = (wg * TILES_PER_WAVE + i) * 16;
        const int w  = w0 + l15;

        // B tiles (4x16 f32): lane l15 = column (w position), half selects K pair
        const int s0 = w * TAU - PADL + kb0;
        const int s1 = w * TAU - PADL + kb1;
        v2f b0, b1;
        b0.x = ldx(xb, s0); b0.y = ldx(xb, s0 + 1);
        b1.x = ldx(xb, s1); b1.y = ldx(xb, s1 + 1);

        // D = A1*B1 + (A0*B0 + 0)  — chained fp32 WMMA, full precision
        v8f acc = {0.f, 0.f, 0.f, 0.f, 0.f, 0.f, 0.f, 0.f};
        acc = __builtin_amdgcn_wmma_f32_16x16x4_f32(
                  false, a0, false, b0, (short)0, acc, false, false);
        acc = __builtin_amdgcn_wmma_f32_16x16x4_f32(
                  false, a1, false, b1, (short)0, acc, false, false);

        // C/D layout: VGPR v holds row (v + 8*half), column = l15
        #pragma unroll
        for (int v = 0; v < 8; ++v) {
            const int n = n0 + v + half * 8;
            int ch;
            if (n < NKER - 1) {
                ch = n * CIN + c;                 // main: 'bwnc' -> channel n*7+c
            } else if (n == NKER - 1 && c == 0) {
                ch = OUTCH - 1;                   // final: channel 511 uses c==0
            } else {
                continue;                         // padded rows 74..79, or n=73 c>0
            }
            ob[(size_t)ch * WOUT + w0 + l15] = acc[v];
        }
    }
}

extern "C" void kernel_launch(void* const* d_in, const int* in_sizes, int n_in,
                              void* d_out, int out_size, void* d_ws, size_t ws_size,
                              hipStream_t stream) {
    (void)in_sizes; (void)n_in; (void)out_size; (void)d_ws; (void)ws_size;
    const float* x    = (const float*)d_in[0];   // (32, 8192, 7) fp32
    const float* kern = (const float*)d_in[1];   // (74, 8) fp32
    float* out        = (float*)d_out;           // (32, 512, 2736) fp32

    dim3 block(WAVES_PER_BLOCK * 32);            // 256 threads = 8 waves
    dim3 grid(WAVES_TOTAL / WAVES_PER_BLOCK);    // 2660 blocks, exact
    rocket_dilconv_wmma<<<grid, block, 0, stream>>>(x, kern, out);
}